// SpectralConvolution_58626303591095
// MI455X (gfx1250) — compile-verified
//
#include <hip/hip_runtime.h>
#include <hip/hip_bf16.h>
#include <math.h>

// ---------------------------------------------------------------------------
// FNO-3D spectral convolution for MI455X (gfx1250).
// Strategy: mode truncation makes every FFT axis a truncated DFT, i.e. a
// small GEMM -> drive everything through V_WMMA_F32_16X16X4_F32 (fp32 WMMA,
// wave32). The problem is HBM-bound (~1.1 GB traffic @ 23.3 TB/s ~ 47 us;
// only ~20 GFLOP of fp32 GEMM), so fp32 WMMA is free and keeps reference
// accuracy.
// ---------------------------------------------------------------------------

typedef __attribute__((ext_vector_type(2))) float v2f;
typedef __attribute__((ext_vector_type(8))) float v8f;

__device__ __forceinline__ v8f wmma4(v2f a, v2f b, v8f c) {
  // D(16x16,f32) = A(16x4,f32) * B(4x16,f32) + C
  return __builtin_amdgcn_wmma_f32_16x16x4_f32(false, a, false, b,
                                               (short)0, c, false, false);
}

#define TWO_PI 6.28318530717958647692f
#define CCH 16   // channels
#define NSP 192  // spatial size
#define MOD 16   // retained modes per axis

// ---- d_ws float offsets -------------------------------------------------
#define OFF_MZF_C 0          // 16x192  z-forward cos
#define OFF_MZF_S 3072       // 16x192  z-forward -sin
#define OFF_MF_C  6144       // 32x192  fwd cos   (shared by y-fwd, x-fwd)
#define OFF_MF_S  12288      // 32x192  fwd +sin
#define OFF_MF_SN 18432      // 32x192  fwd -sin
#define OFF_MI_C  24576      // 192x32  inv cos/192
#define OFF_MI_S  30720      // 192x32  inv -sin/192
#define OFF_MI_SN 36864      // 192x32  inv +sin/192
#define OFF_MZI   43008      // 192x32  z-inverse real matrix
#define OFF_BUF1  49152ull       // (16,192,192,16) cplx = 18,874,368 f  (Xz / Z2)
#define OFF_BUF2  18923520ull    // (16,192, 32,16) cplx =  3,145,728 f  (Xy / Z1)
#define OFF_BUF3  22069248ull    // (16, 32, 32,16) cplx =    524,288 f  (Xx)
#define OFF_BUF4  22593536ull    // (16, 32, 32,16) cplx =    524,288 f  (Y)
// total = 23,117,824 floats ~ 92.5 MB

// ---------------------------------------------------------------------------
// Build all DFT matrices. Angles reduced mod 192 in *integer* arithmetic
// before sinf/cosf so large freq*index products stay exact.
// ---------------------------------------------------------------------------
__global__ void build_mats(float* __restrict__ ws) {
  int i = blockIdx.x * blockDim.x + threadIdx.x;
  if (i < 3072) {                       // z-forward: 16 modes x 192 samples
    int m = i / 192, k = i % 192;
    float th = (TWO_PI / 192.0f) * (float)((m * k) % 192);
    ws[OFF_MZF_C + i] = cosf(th);
    ws[OFF_MZF_S + i] = -sinf(th);      // e^{-i th} on real input
  } else if (i < 9216) {                // fwd 32 freqs {0..15,176..191} x 192
    int j = i - 3072;
    int m = j / 192, k = j % 192;
    int f = (m < 16) ? m : (160 + m);   // 176 + (m-16)
    float th = (TWO_PI / 192.0f) * (float)((f * k) % 192);
    float c = cosf(th), s = sinf(th);
    ws[OFF_MF_C + j] = c;               // re = C*xr + S*xi ; im = C*xi - S*xr
    ws[OFF_MF_S + j] = s;
    ws[OFF_MF_SN + j] = -s;
  } else if (i < 15360) {               // inverse: 192 positions x 32 freqs
    int j = i - 9216;
    int n = j / 32, k = j % 32;
    int f = (k < 16) ? k : (160 + k);
    float th = (TWO_PI / 192.0f) * (float)((f * n) % 192);
    const float sc = 1.0f / 192.0f;     // e^{+i th}/192
    ws[OFF_MI_C + j] = sc * cosf(th);   // re = C*xr - S*xi ; im = C*xi + S*xr
    ws[OFF_MI_S + j] = -sc * sinf(th);
    ws[OFF_MI_SN + j] = sc * sinf(th);
  } else if (i < 21504) {               // z-inverse: 192 x (16 cplx as 32 real)
    int j = i - 15360;
    int n = j / 32, kp = j % 32, k = kp >> 1;
    float a = ((k == 0) ? 1.0f : 2.0f) / 192.0f;  // irfft Hermitian fold
    float th = (TWO_PI / 192.0f) * (float)((k * n) % 192);
    ws[OFF_MZI + j] = (kp & 1) ? (-a * sinf(th)) : (a * cosf(th));
  }
}

// ---------------------------------------------------------------------------
// Stage 1: z-forward truncated DFT. real (line,192) -> complex (line,16).
// One wave computes a 16 modes x 16 lines tile; K=192 in steps of 4.
// B loads are contiguous float2 (z is innermost in x).
// ---------------------------------------------------------------------------
__global__ void dft_zfwd(const float* __restrict__ x, float* __restrict__ out,
                         const float* __restrict__ Ac,
                         const float* __restrict__ As) {
  const int lane = threadIdx.x & 31;
  const int wave = threadIdx.x >> 5;
  const int job = blockIdx.x * (blockDim.x >> 5) + wave;
  const int njobs = (CCH * NSP * NSP) >> 4;  // 36864 waves
  if (job >= njobs) return;
  const int row = lane & 15;
  const int hi = lane >> 4;                  // K-half select
  const size_t linebase = (size_t)job << 4;
  const float* bp = x + (linebase + row) * NSP;   // lane's line (B column)
  const float* acp = Ac + row * NSP;              // lane's mode row (A)
  const float* asp = As + row * NSP;
  v8f aR = {}; v8f aI = {};
#pragma unroll 4
  for (int K0 = 0; K0 < NSP; K0 += 4) {
    const int kb = K0 + 2 * hi;
    v2f a1 = *(const v2f*)(acp + kb);
    v2f a2 = *(const v2f*)(asp + kb);
    v2f b  = *(const v2f*)(bp + kb);
    aR = wmma4(a1, b, aR);
    aI = wmma4(a2, b, aI);
  }
  // D layout: lane -> N (line), vgpr r -> M = 8*hi + r (mode). Complex out.
  float* op = out + ((linebase + row) * MOD + 8 * hi) * 2;
#pragma unroll
  for (int r = 0; r < 8; ++r) {
    v2f v = {aR[r], aI[r]};
    *(v2f*)(op + r * 2) = v;
  }
}

// ---------------------------------------------------------------------------
// Generic complex DFT-GEMM over one axis.
// Data viewed as (O, K, I) complex, I contiguous. out is (O, M, I) complex.
//   re = Ac*xr + As*xi ;  im = Ac*xi + Asn*xr     (signs baked into matrices)
// One wave = one (o, 16-mode tile, 16-line tile). 4 WMMAs per K-step.
// ---------------------------------------------------------------------------
__global__ void dft_cgemm(const float* __restrict__ in, float* __restrict__ out,
                          const float* __restrict__ Ac,
                          const float* __restrict__ As,
                          const float* __restrict__ Asn,
                          int O, int K, int I, int M) {
  const int lane = threadIdx.x & 31;
  const int wave = threadIdx.x >> 5;
  const int mt = M >> 4, it = I >> 4;
  const int job = blockIdx.x * (blockDim.x >> 5) + wave;
  const int njobs = O * mt * it;
  if (job >= njobs) return;
  const int o = job / (mt * it);
  const int r2 = job - o * (mt * it);
  const int mtile = r2 / it;
  const int itile = r2 - mtile * it;
  const int row = lane & 15;
  const int hi = lane >> 4;

  const float* acp = Ac + (size_t)(mtile * 16 + row) * K;
  const float* asp = As + (size_t)(mtile * 16 + row) * K;
  const float* anp = Asn + (size_t)(mtile * 16 + row) * K;
  const float* bp = in + ((size_t)o * K * I + itile * 16 + row) * 2;
  const size_t kstr = (size_t)I * 2;  // float stride between K rows

  v8f accR = {}; v8f accI = {};
  for (int K0 = 0; K0 < K; K0 += 4) {
    const int kb = K0 + 2 * hi;
    v2f ac = *(const v2f*)(acp + kb);
    v2f as = *(const v2f*)(asp + kb);
    v2f an = *(const v2f*)(anp + kb);
    v2f x0 = *(const v2f*)(bp + (size_t)kb * kstr);        // (re,im) at k
    v2f x1 = *(const v2f*)(bp + (size_t)(kb + 1) * kstr);  // (re,im) at k+1
    v2f br = {x0.x, x1.x};
    v2f bi = {x0.y, x1.y};
    accR = wmma4(ac, br, accR);
    accR = wmma4(as, bi, accR);
    accI = wmma4(ac, bi, accI);
    accI = wmma4(an, br, accI);
  }
  // out[(o*M + m)*I + n], m = mtile*16 + 8*hi + r, n = itile*16 + row
  size_t base = (((size_t)o * M + mtile * 16 + 8 * hi) * I + itile * 16 + row) * 2;
#pragma unroll
  for (int r = 0; r < 8; ++r) {
    v2f v = {accR[r], accI[r]};
    *(v2f*)(out + base + (size_t)r * kstr) = v;
  }
}

// ---------------------------------------------------------------------------
// Per-mode 16x16 complex channel mix (einsum 'ixyz,ioxyz->oxyz' per corner).
// Weights differ per mode -> not GEMM-able; 33 MFLOP, plain VALU.
// Spectrum layout: (c, kx32, ky32, kz16); tile index >=16 <-> freqs 176..191,
// so corner = (kx>=16) + 2*(ky>=16) matches w[0..3] block order exactly.
// ---------------------------------------------------------------------------
__global__ void mode_mix(const float* __restrict__ X, const float* __restrict__ wr,
                         const float* __restrict__ wi, float* __restrict__ Y) {
  const int idx = blockIdx.x * blockDim.x + threadIdx.x;  // 16*32*32*16
  if (idx >= CCH * 32 * 32 * MOD) return;
  const int kz = idx & 15;
  const int ky = (idx >> 4) & 31;
  const int kx = (idx >> 9) & 31;
  const int co = idx >> 14;
  const int corner = (kx >> 4) | ((ky >> 4) << 1);
  const int mode = (kx & 15) * 256 + (ky & 15) * 16 + kz;  // w[..., x,y,z]
  const size_t xoff = (((size_t)kx * 32 + ky) * MOD + kz) * 2;
  float accr = 0.f, acci = 0.f;
#pragma unroll 4
  for (int ci = 0; ci < CCH; ++ci) {
    const float xr = X[xoff + (size_t)ci * 32768];
    const float xi = X[xoff + (size_t)ci * 32768 + 1];
    const size_t w = ((size_t)(corner * 16 + ci) * 16 + co) * 4096 + mode;
    const float a = wr[w], b = wi[w];
    accr += a * xr - b * xi;   // (xr+i xi)(a+i b)
    acci += a * xi + b * xr;
  }
  const size_t yoff = (((size_t)co * 32 + kx) * 32 + ky) * MOD * 2 + kz * 2;
  Y[yoff] = accr;
  Y[yoff + 1] = acci;
}

// ---------------------------------------------------------------------------
// z-inverse: real-output C2R. The 16 complex kz modes are 32 contiguous
// reals, so this is a single *real* GEMM: out(16y x 16z) += H(16z x 32) * B.
// ---------------------------------------------------------------------------
__global__ void dft_zinv(const float* __restrict__ in, float* __restrict__ out,
                         const float* __restrict__ H) {
  const int lane = threadIdx.x & 31;
  const int wave = threadIdx.x >> 5;
  const int job = blockIdx.x * (blockDim.x >> 5) + wave;
  const int njobs = CCH * NSP * 12 * 12;  // 442368 waves
  if (job >= njobs) return;
  const int o = job / 144;            // (c,x)
  const int r2 = job - o * 144;
  const int yt = r2 / 12, zt = r2 - yt * 12;
  const int row = lane & 15;
  const int hi = lane >> 4;
  const float* bp = in + ((size_t)o * NSP + yt * 16 + row) * 32;  // line = y
  const float* hp = H + (size_t)(zt * 16 + row) * 32;             // A row = z
  v8f acc = {};
#pragma unroll
  for (int K0 = 0; K0 < 32; K0 += 4) {
    const int kb = K0 + 2 * hi;
    v2f a = *(const v2f*)(hp + kb);
    v2f b = *(const v2f*)(bp + kb);
    acc = wmma4(a, b, acc);
  }
  float* op = out + ((size_t)o * NSP + yt * 16 + row) * NSP + zt * 16 + 8 * hi;
#pragma unroll
  for (int j = 0; j < 4; ++j) {
    v2f v = {acc[2 * j], acc[2 * j + 1]};
    *(v2f*)(op + 2 * j) = v;  // 8 consecutive z per lane, as 4x float2
  }
}

// ---------------------------------------------------------------------------
extern "C" void kernel_launch(void* const* d_in, const int* in_sizes, int n_in,
                              void* d_out, int out_size, void* d_ws, size_t ws_size,
                              hipStream_t stream) {
  (void)in_sizes; (void)n_in; (void)out_size; (void)ws_size;
  const float* x = (const float*)d_in[0];
  const float* wr = (const float*)d_in[1];
  const float* wi = (const float*)d_in[2];
  // d_in[3] = modes (int, ==16) is compiled in as MOD.
  float* out = (float*)d_out;
  float* ws = (float*)d_ws;

  float* MZF_C = ws + OFF_MZF_C;
  float* MZF_S = ws + OFF_MZF_S;
  float* MF_C = ws + OFF_MF_C;
  float* MF_S = ws + OFF_MF_S;
  float* MF_SN = ws + OFF_MF_SN;
  float* MI_C = ws + OFF_MI_C;
  float* MI_S = ws + OFF_MI_S;
  float* MI_SN = ws + OFF_MI_SN;
  float* MZI = ws + OFF_MZI;
  float* BUF1 = ws + OFF_BUF1;  // Xz, later Z2
  float* BUF2 = ws + OFF_BUF2;  // Xy, later Z1
  float* BUF3 = ws + OFF_BUF3;  // Xx
  float* BUF4 = ws + OFF_BUF4;  // Y (mixed spectrum)

  // 0) twiddle matrices (rebuilt every call; no cached state)
  build_mats<<<(21504 + 255) / 256, 256, 0, stream>>>(ws);
  // 1) z-forward: x (16,192,192,192) -> Xz (16,192,192,16) cplx
  dft_zfwd<<<36864 / 8, 256, 0, stream>>>(x, BUF1, MZF_C, MZF_S);
  // 2) y-forward: O=(c,x)=3072, K=192, I=16, M=32 -> Xy (16,192,32,16)
  dft_cgemm<<<6144 / 8, 256, 0, stream>>>(BUF1, BUF2, MF_C, MF_S, MF_SN,
                                          3072, 192, 16, 32);
  // 3) x-forward: O=c=16, K=192, I=512, M=32 -> Xx (16,32,32,16)
  dft_cgemm<<<1024 / 8, 256, 0, stream>>>(BUF2, BUF3, MF_C, MF_S, MF_SN,
                                          16, 192, 512, 32);
  // 4) per-mode channel mix -> Y (16,32,32,16)
  mode_mix<<<262144 / 256, 256, 0, stream>>>(BUF3, wr, wi, BUF4);
  // 5) x-inverse: O=16, K=32, I=512, M=192 -> Z1 (16,192,32,16)
  dft_cgemm<<<6144 / 8, 256, 0, stream>>>(BUF4, BUF2, MI_C, MI_S, MI_SN,
                                          16, 32, 512, 192);
  // 6) y-inverse: O=3072, K=32, I=16, M=192 -> Z2 (16,192,192,16)
  dft_cgemm<<<36864 / 8, 256, 0, stream>>>(BUF2, BUF1, MI_C, MI_S, MI_SN,
                                           3072, 32, 16, 192);
  // 7) z-inverse C2R -> out (16,192,192,192)
  dft_zinv<<<442368 / 8, 256, 0, stream>>>(BUF1, out, MZI);
}